// ModulatedConv2D_58892591563242
// MI455X (gfx1250) — compile-verified
//
#include <hip/hip_runtime.h>

typedef __attribute__((ext_vector_type(2))) float v2f;
typedef __attribute__((ext_vector_type(8))) float v8f;

#define B_    8
#define CIN_  512
#define COUT_ 3
#define H_    256
#define W_    256
#define HW_   (H_ * W_)
#define ZDIM_ 512
#define WPAD_ 16   // padded cout stride for the modulated weight matrix

// ---------------------------------------------------------------------------
// Kernel 1: style dense + modulate + demodulate.
// One block per batch, one thread per cin. Deterministic tree reduction for
// the demodulation sum-of-squares (no FP atomics -> bit-stable every replay).
// Writes wmodT[b][cin][0..15] (cols 3..15 zeroed) into workspace.
// ---------------------------------------------------------------------------
__global__ __launch_bounds__(CIN_)
void modstyle_kernel(const float* __restrict__ y,
                     const float* __restrict__ w,
                     const float* __restrict__ dense_w,
                     const float* __restrict__ dense_b,
                     float* __restrict__ wmodT) {
  const int b   = blockIdx.x;
  const int cin = threadIdx.x;

  __shared__ float yb[ZDIM_];
  __shared__ float red[CIN_];
  __shared__ float inv_d[COUT_];

  yb[cin] = y[b * ZDIM_ + cin];
  __syncthreads();

  float acc = 0.f;
  #pragma unroll 8
  for (int z = 0; z < ZDIM_; ++z)
    acc = fmaf(yb[z], dense_w[z * CIN_ + cin], acc);

  const float rc_dense = 0.04419417382415922f;  // 1/sqrt(512)
  const float rc_conv  = 0.04419417382415922f;  // 1/sqrt(1*1*512)
  const float s = acc * rc_dense + dense_b[cin] + 1.0f;

  float ww[COUT_];
  #pragma unroll
  for (int c = 0; c < COUT_; ++c)
    ww[c] = rc_conv * w[cin * COUT_ + c] * s;

  #pragma unroll
  for (int c = 0; c < COUT_; ++c) {
    red[cin] = ww[c] * ww[c];
    __syncthreads();
    for (int stride = CIN_ / 2; stride > 0; stride >>= 1) {
      if (cin < stride) red[cin] += red[cin + stride];
      __syncthreads();
    }
    if (cin == 0) inv_d[c] = rsqrtf(red[0] + 1e-8f);
    __syncthreads();
  }

  float* dst = wmodT + ((size_t)b * CIN_ + cin) * WPAD_;
  #pragma unroll
  for (int c = 0; c < COUT_; ++c) dst[c] = ww[c] * inv_d[c];
  #pragma unroll
  for (int c = COUT_; c < WPAD_; ++c) dst[c] = 0.f;
}

// ---------------------------------------------------------------------------
// Kernel 2: per-batch 1x1 conv as WMMA f32 16x16x4 GEMM.
//   D[pix, cout] += A[pix, k] * B[k, cout]
// A comes straight from NCHW x (coalesced, non-temporal: 1 GiB streamed once),
// B from LDS (32 KB per block, loaded once). 4 pixel tiles per wave.
// ---------------------------------------------------------------------------
__global__ __launch_bounds__(256)
void modconv_wmma_kernel(const float* __restrict__ x,
                         const float* __restrict__ wmodT,
                         float* __restrict__ out) {
  __shared__ float sW[CIN_ * WPAD_];  // 32 KB

  const int b   = blockIdx.y;
  const int tid = threadIdx.x;

  // Cooperative fill of this batch's padded weights (8192 floats, b128 loads).
  {
    const float4* src = (const float4*)(wmodT + (size_t)b * CIN_ * WPAD_);
    float4*       dst = (float4*)sW;
    #pragma unroll
    for (int i = 0; i < (CIN_ * WPAD_ / 4) / 256; ++i)
      dst[tid + i * 256] = src[tid + i * 256];
  }
  __syncthreads();

  const int wave = tid >> 5;
  const int lane = tid & 31;
  const int half = lane >> 4;   // 0: K = k0,k0+1   1: K = k0+2,k0+3
  const int l15  = lane & 15;   // A: pixel-in-tile / B: cout column
  const int pix0 = blockIdx.x * 512 + wave * 64;

  const float* xb = x + (size_t)b * CIN_ * HW_ + pix0 + l15;

  v8f acc0 = {}, acc1 = {}, acc2 = {}, acc3 = {};

  for (int k0 = 0; k0 < CIN_; k0 += 4) {
    const int kr = k0 + 2 * half;

    const float* wp = sW + kr * WPAD_ + l15;
    v2f bv;
    bv.x = wp[0];
    bv.y = wp[WPAD_];

    const float* xp = xb + (size_t)kr * HW_;
    v2f a;

    a.x = __builtin_nontemporal_load(xp);
    a.y = __builtin_nontemporal_load(xp + HW_);
    acc0 = __builtin_amdgcn_wmma_f32_16x16x4_f32(false, a, false, bv,
                                                 (short)0, acc0, false, false);

    a.x = __builtin_nontemporal_load(xp + 16);
    a.y = __builtin_nontemporal_load(xp + HW_ + 16);
    acc1 = __builtin_amdgcn_wmma_f32_16x16x4_f32(false, a, false, bv,
                                                 (short)0, acc1, false, false);

    a.x = __builtin_nontemporal_load(xp + 32);
    a.y = __builtin_nontemporal_load(xp + HW_ + 32);
    acc2 = __builtin_amdgcn_wmma_f32_16x16x4_f32(false, a, false, bv,
                                                 (short)0, acc2, false, false);

    a.x = __builtin_nontemporal_load(xp + 48);
    a.y = __builtin_nontemporal_load(xp + HW_ + 48);
    acc3 = __builtin_amdgcn_wmma_f32_16x16x4_f32(false, a, false, bv,
                                                 (short)0, acc3, false, false);
  }

  // D layout: VGPR r -> pixel (r + 8*half) within the 16-pixel tile,
  // lane&15 -> cout column. Only cout < 3 lanes store.
  if (l15 < COUT_) {
    float* op = out + ((size_t)b * COUT_ + l15) * HW_ + pix0 + 8 * half;
    #pragma unroll
    for (int r = 0; r < 8; ++r) {
      __builtin_nontemporal_store(acc0[r], op + r);
      __builtin_nontemporal_store(acc1[r], op + 16 + r);
      __builtin_nontemporal_store(acc2[r], op + 32 + r);
      __builtin_nontemporal_store(acc3[r], op + 48 + r);
    }
  }
}

// ---------------------------------------------------------------------------
extern "C" void kernel_launch(void* const* d_in, const int* in_sizes, int n_in,
                              void* d_out, int out_size, void* d_ws, size_t ws_size,
                              hipStream_t stream) {
  const float* x       = (const float*)d_in[0];  // [8, 512, 256, 256]
  const float* y       = (const float*)d_in[1];  // [8, 512]
  const float* w       = (const float*)d_in[2];  // [1, 1, 512, 3]
  const float* dense_w = (const float*)d_in[3];  // [512, 512]
  const float* dense_b = (const float*)d_in[4];  // [512]
  float*       out     = (float*)d_out;          // [8, 3, 256, 256]
  float*       wmodT   = (float*)d_ws;           // [8][512][16] = 256 KB

  modstyle_kernel<<<B_, CIN_, 0, stream>>>(y, w, dense_w, dense_b, wmodT);

  dim3 grid(HW_ / 512, B_);
  modconv_wmma_kernel<<<grid, 256, 0, stream>>>(x, wmodT, out);
}